// QuadraticModule_5068061409905
// MI455X (gfx1250) — compile-verified
//
#include <hip/hip_runtime.h>

// DennisNode Q-module update, MI455X / gfx1250 (wave32).
// Memory-bound streaming kernel (136 MB @ 23.3 TB/s ~= 5.9 us floor).
// Coupling matvec done with V_WMMA_F32_16X16X4_F32 (fp32 == reference math):
//   D = A(16x4 coupling, zero-padded) x B(4x16 = Z_tile^T)  ->  z_nb
// B operand is loaded straight from global memory in WMMA B layout
// (lane n<16: Z[base+n].xy ; lane n+16: Z[base+n].zw), no shuffles needed.
// Results redistributed to one-lane-per-instance via ds_bpermute_b32.

typedef float v2f __attribute__((ext_vector_type(2)));
typedef float v4f __attribute__((ext_vector_type(4)));
typedef float v8f __attribute__((ext_vector_type(8)));

#define K_GAMMA   0.103f
#define K_DT      0.01f
#define K_OMEGA   1.0f
#define K_ENTRAIN 0.3f
#define K_AMP     0.5f
#define K_DAMP    0.1f

#if __has_builtin(__builtin_amdgcn_wmma_f32_16x16x4_f32)
#define USE_WMMA_K4 1
#else
#define USE_WMMA_K4 0
#endif

__global__ __launch_bounds__(256)
void dennis_q_update(const float* __restrict__ Z,
                     const float* __restrict__ phase,
                     const float* __restrict__ ext,
                     const float* __restrict__ dphase,
                     const float* __restrict__ coupling,
                     float* __restrict__ out,
                     int nB)
{
    const int lane = threadIdx.x & 31;
    const int wave = blockIdx.x * (blockDim.x >> 5) + (threadIdx.x >> 5);
    const int base = wave << 5;            // first instance owned by this wave
    // nB % 32 == 0 (B = 2,000,000), so active waves own a full 32-instance
    // tile and this early-out is wave-uniform -> EXEC all-1s at the WMMAs.
    if (base >= nB) return;

    const int  row    = lane & 15;
    const bool hiHalf = lane >= 16;
    const int  g      = base + lane;       // this lane's instance (elementwise part)

    float nb[4];                           // z_nb[g][0..3] = coupling @ Z[g]

#if USE_WMMA_K4
    // ---- A operand: coupling (4x4) zero-padded to 16x4, documented A layout:
    // lane M (0-15): v0=A[M][0], v1=A[M][1]; lanes 16-31: v0=A[M][2], v1=A[M][3]
    v2f a = {0.0f, 0.0f};
    if (row < 4)
        a = *(const v2f*)(coupling + row * 4 + (hiHalf ? 2 : 0));

    // ---- B operands: Z_tile^T (4x16), loaded directly in B layout.
    // lane n (0-15): v0=Z[base+n][0], v1=Z[base+n][1]
    // lane n+16    : v0=Z[base+n][2], v1=Z[base+n][3]
    v2f b0 = *(const v2f*)(Z + (base      + row) * 4 + (hiHalf ? 2 : 0));
    v2f b1 = *(const v2f*)(Z + (base + 16 + row) * 4 + (hiHalf ? 2 : 0));

    v8f cz = {};
    // D[m][n] = sum_k coupling[m][k] * Z[base+n][k] = z_nb[base+n][m]
    v8f d0 = __builtin_amdgcn_wmma_f32_16x16x4_f32(
        false, a, false, b0, (short)0, cz, false, false);
    v8f d1 = __builtin_amdgcn_wmma_f32_16x16x4_f32(
        false, a, false, b1, (short)0, cz, false, false);
    // D layout: VGPR i, lanes 0-15 -> M=i, N=lane. So lanes 0-15 hold
    // z_nb[(base|base+16)+n][i] in d{0,1}[i], i = 0..3. Lanes 16-31 / VGPRs
    // 4-7 are the zero-padded rows (ignored).

    // ---- redistribute to one-lane-per-instance via ds_bpermute ----
    const int pidx = row << 2;             // byte index: gather from lane (lane&15)
    #pragma unroll
    for (int i = 0; i < 4; ++i) {
        int lo = __builtin_amdgcn_ds_bpermute(pidx, __float_as_int(d0[i]));
        int hi = __builtin_amdgcn_ds_bpermute(pidx, __float_as_int(d1[i]));
        nb[i]  = __int_as_float(hiHalf ? hi : lo);
    }
#else
    // Fallback: per-lane 4x4 matvec with FMAs (coupling broadcast, L0-cached).
    {
        const v4f zv = *(const v4f*)(Z + g * 4);
        #pragma unroll
        for (int i = 0; i < 4; ++i) {
            const v4f ci = *(const v4f*)(coupling + i * 4);
            nb[i] = ci[0] * zv[0] + ci[1] * zv[1] + ci[2] * zv[2] + ci[3] * zv[3];
        }
    }
#endif

    // ---- elementwise DennisNode update, one instance per lane ----
    const v4f   zv = *(const v4f*)(Z     + g * 4);   // hits WGP$ (just streamed)
    const v4f   pv = *(const v4f*)(phase + g * 4);
    const v4f   ev = *(const v4f*)(ext   + g * 4);
    const float dp = dphase[g];

    v4f zo;
    #pragma unroll
    for (int i = 0; i < 4; ++i) {
        float znb = nb[i] + 0.1f * ev[i];
        float np  = pv[i] + K_DT * K_OMEGA + K_ENTRAIN * __sinf(dp - pv[i]);
        float v   = zv[i] + K_DT * (znb + K_AMP * __cosf(np) - K_DAMP * zv[i]);
        zo[i] = fminf(K_GAMMA, fmaxf(-K_GAMMA, v));
    }
    *(v4f*)(out + g * 4) = zo;             // coalesced b128 store
}

extern "C" void kernel_launch(void* const* d_in, const int* in_sizes, int n_in,
                              void* d_out, int out_size, void* d_ws, size_t ws_size,
                              hipStream_t stream) {
    const float* Z        = (const float*)d_in[0];   // [B,4]
    const float* phase    = (const float*)d_in[1];   // [B,4]
    const float* ext      = (const float*)d_in[2];   // [B,4]
    const float* dphase   = (const float*)d_in[3];   // [B,1]
    const float* coupling = (const float*)d_in[4];   // [4,4]
    float* out = (float*)d_out;                      // [B,4]

    const int B = in_sizes[0] / 4;
    // 256 threads = 8 waves = 256 instances per block
    const int blocks = (B + 255) / 256;
    dennis_q_update<<<blocks, 256, 0, stream>>>(Z, phase, ext, dphase,
                                                coupling, out, B);
}